// NetVLAD_65635690217959
// MI455X (gfx1250) — compile-verified
//
#include <hip/hip_runtime.h>
#include <hip/hip_bf16.h>
#include <math.h>

// Problem constants (from reference setup_inputs)
#define B_  16
#define D_  512
#define N_  4096      // H*W = 64*64
#define K_  64
#define TN  64        // spatial positions per tile
#define TILES_PER_BLK 4
#define BLKS_PER_BATCH 16           // 64 tiles / 4 per block
#define XS  68        // xn LDS row stride (floats): 272B rows -> 16B aligned, odd bank phase
#define AS  66        // assign LDS row stride: 8B-aligned pairs, conflict-free column reads
#define XBUF (D_ * XS)              // one tile buffer, in floats
#define EPSV 1e-12f

typedef float v2f __attribute__((ext_vector_type(2)));
typedef float v8f __attribute__((ext_vector_type(8)));

// --- CDNA5 async global->LDS path (guarded; falls back to plain LDS stores) ---
#ifndef ASYNC_TRY
#define ASYNC_TRY 1
#endif
#if ASYNC_TRY && defined(__has_builtin)
#if __has_builtin(__builtin_amdgcn_global_load_async_to_lds_b128)
#define HAVE_ASYNC 1
#endif
#endif

#if defined(HAVE_ASYNC)
// Probed signature: param0 = `int __vector(4) __device__ *` (AS1), param1 = AS3.
typedef int v4i_vs __attribute__((vector_size(16)));
typedef __attribute__((address_space(1))) v4i_vs* gas_b128_t;
typedef __attribute__((address_space(3))) v4i_vs* las_b128_t;
#endif

// V_WMMA_F32_16X16X4_F32: D(16x16 f32, 8 VGPR) = A(16x4 f32, 2 VGPR) * B(4x16 f32, 2 VGPR) + C
__device__ __forceinline__ v8f wmma4(v2f a, v2f b, v8f c) {
  return __builtin_amdgcn_wmma_f32_16x16x4_f32(false, a, false, b, (short)0, c,
                                               false, false);
}

__global__ void zero_ws_kernel(float* __restrict__ p, int n) {
  int i = blockIdx.x * blockDim.x + threadIdx.x;
  if (i < n) p[i] = 0.0f;
}

// Issue one tile's worth (512 x 64 floats) of async b128 global->LDS transfers.
// 32 wave-level async instructions per wave (ASYNCcnt += 32).
// NOTE: all LDS addressing is smem[base + off] so address-space inference keeps
// these (and all compute-loop accesses) on the DS/async path, not FLAT.
__device__ __forceinline__ void stage_tile_async(const float* __restrict__ xb,
                                                 float* smem, int off, int tid) {
#if defined(HAVE_ASYNC)
  for (int idx = tid; idx < D_ * (TN / 4); idx += 256) {
    const int d = idx >> 4, q = (idx & 15) * 4;
    float* gp = const_cast<float*>(xb + (size_t)d * N_ + q);
    float* lp = &smem[off + d * XS + q];
    __builtin_amdgcn_global_load_async_to_lds_b128(
        (gas_b128_t)gp, (las_b128_t)lp, 0, 0);
  }
#endif
}

__device__ __forceinline__ void stage_tile_sync(const float* __restrict__ xb,
                                                float* smem, int off, int tid) {
  for (int idx = tid; idx < D_ * (TN / 4); idx += 256) {
    const int d = idx >> 4, q = (idx & 15) * 4;
    const float4 v = *(const float4*)(xb + (size_t)d * N_ + q);
    float* lp = &smem[off + d * XS + q];
    lp[0] = v.x; lp[1] = v.y; lp[2] = v.z; lp[3] = v.w;
  }
}

// Fused: normalize(x) -> logits = W@xn + b -> softmax_K -> agg += assign@xn^T, mass += sum_n
__global__ void __launch_bounds__(256)
netvlad_main_kernel(const float* __restrict__ x,
                    const float* __restrict__ conv_w,
                    const float* __restrict__ conv_b,
                    float* __restrict__ agg,    // [B,K,D] fp32, pre-zeroed
                    float* __restrict__ mass)   // [B,K]   fp32, pre-zeroed
{
  extern __shared__ float smem[];
  // smem[0 .. XBUF)        : xn tile buffer 0   [D_][XS]
  // smem[XBUF .. 2*XBUF)   : xn tile buffer 1   [D_][XS]
  float* asg  = smem + 2 * XBUF;      // [K_][AS] logits -> softmax assignments
  float* invn = asg + K_ * AS;        // [TN]     per-position 1/||x||

  const int tid  = threadIdx.x;
  const int lane = tid & 31;
  const int wave = tid >> 5;          // 8 waves of 32 (wave32)
  const int half = lane >> 4;
  const int lm   = lane & 15;

  const int b   = blockIdx.x / BLKS_PER_BATCH;
  const int blk = blockIdx.x % BLKS_PER_BATCH;
  const float* xbatch = x + ((size_t)b * D_) * (size_t)N_;

  // GEMM1: wave covers tiles (mi1, ni1b) and (mi1, ni1b+1) -> shared A fragment
  const int mi1  = wave >> 1;
  const int ni1b = (wave & 1) * 2;
  // GEMM2 persistent tiles: cluster-tile mi2, d-tiles [ni2b, ni2b+16)
  const int mi2  = wave >> 1;
  const int ni2b = (wave & 1) * 16;
  v8f acc2[16];
#pragma unroll
  for (int i = 0; i < 16; ++i) acc2[i] = v8f{};

#if defined(HAVE_ASYNC)
  // Prefetch tile 0 into buffer 0.
  stage_tile_async(xbatch + (size_t)(blk * TILES_PER_BLK) * TN, smem, 0, tid);
#endif

  for (int t = 0; t < TILES_PER_BLK; ++t) {
    const int xo = (t & 1) * XBUF;    // integer offset keeps accesses on DS path
#if defined(HAVE_ASYNC)
    if (t + 1 < TILES_PER_BLK) {
      // Prefetch next tile into the other buffer, then wait only for the
      // current tile's 32 in-flight transfers (async loads complete in order).
      stage_tile_async(xbatch + (size_t)(blk * TILES_PER_BLK + t + 1) * TN,
                       smem, XBUF - xo, tid);
      asm volatile("s_wait_asynccnt 0x20" ::: "memory");
    } else {
      asm volatile("s_wait_asynccnt 0x0" ::: "memory");
    }
#else
    stage_tile_sync(xbatch + (size_t)(blk * TILES_PER_BLK + t) * TN, smem, xo, tid);
#endif
    __syncthreads();

    // ---- Per-position L2 norm over D ----
    if (tid < TN) {
      float ss = 0.0f;
      for (int d = 0; d < D_; ++d) { float v = smem[xo + d * XS + tid]; ss += v * v; }
      invn[tid] = 1.0f / fmaxf(sqrtf(ss), EPSV);
    }
    __syncthreads();
    for (int idx = tid; idx < D_ * TN; idx += 256) {
      const int d = idx >> 6, n = idx & 63;
      smem[xo + d * XS + n] *= invn[n];
    }
    __syncthreads();

    // ---- GEMM1: logits(64x64) = conv_w(64x512) @ xn(512x64) + conv_b ----
    // Two 16x16 tiles per wave sharing the A fragment; dual accumulator chains.
    {
      v8f accA = v8f{}, accB = v8f{};
      const float* wrow = conv_w + (size_t)(mi1 * 16 + lm) * D_ + 2 * half;
      const int nbA = xo + (ni1b    ) * 16 + lm;
      const int nbB = xo + (ni1b + 1) * 16 + lm;
      for (int d0 = 0; d0 < D_; d0 += 4) {
        v2f a = *(const v2f*)(wrow + d0);                 // A[m, d0+2h .. +1]
        const int r0 = (d0 + 2 * half) * XS;
        v2f b0, b1;
        b0.x = smem[r0 + nbA];      b0.y = smem[r0 + XS + nbA];
        b1.x = smem[r0 + nbB];      b1.y = smem[r0 + XS + nbB];
        accA = wmma4(a, b0, accA);
        accB = wmma4(a, b1, accB);
      }
#pragma unroll
      for (int r = 0; r < 8; ++r) {
        const int m = mi1 * 16 + r + 8 * half;
        const float bias = conv_b[m];
        asg[m * AS + (ni1b    ) * 16 + lm] = accA[r] + bias;
        asg[m * AS + (ni1b + 1) * 16 + lm] = accB[r] + bias;
      }
    }
    __syncthreads();

    // ---- Softmax over K per position ----
    if (tid < TN) {
      const int n = tid;
      float mx = -1e30f;
      for (int m = 0; m < K_; ++m) mx = fmaxf(mx, asg[m * AS + n]);
      float s = 0.0f;
      for (int m = 0; m < K_; ++m) {
        float e = __expf(asg[m * AS + n] - mx);
        asg[m * AS + n] = e;
        s += e;
      }
      const float inv = 1.0f / s;
      for (int m = 0; m < K_; ++m) asg[m * AS + n] *= inv;
    }
    __syncthreads();

    // ---- Cluster mass for this tile ----
    if (tid < K_) {
      float s = 0.0f;
      for (int n = 0; n < TN; ++n) s += asg[tid * AS + n];
      atomicAdd(&mass[b * K_ + tid], s);
    }

    // ---- GEMM2: acc2 += assign(64x64) @ xn^T(64x512) ----
    // A fragment shared by all 16 tiles of the wave; 16 independent WMMA chains.
    for (int kk = 0; kk < TN; kk += 4) {
      const v2f a = *(const v2f*)&asg[(mi2 * 16 + lm) * AS + kk + 2 * half];
#pragma unroll
      for (int i = 0; i < 16; ++i) {
        const int dcol = (ni2b + i) * 16 + lm;
        const v2f bb = *(const v2f*)&smem[xo + dcol * XS + kk + 2 * half];
        acc2[i] = wmma4(a, bb, acc2[i]);
      }
    }
    __syncthreads();   // protect LDS reuse next tile iteration
  }

  // ---- Flush GEMM2 accumulators to global agg with f32 atomics ----
#pragma unroll
  for (int i = 0; i < 16; ++i) {
    const int dcol = (ni2b + i) * 16 + lm;
#pragma unroll
    for (int r = 0; r < 8; ++r) {
      const int k = mi2 * 16 + r + 8 * half;
      atomicAdd(&agg[((size_t)b * K_ + k) * D_ + dcol], acc2[i][r]);
    }
  }
}

// vlad = agg - mass*c ; intra-normalize over D ; accumulate per-batch sum of squares
__global__ void __launch_bounds__(256)
netvlad_finalize_kernel(const float* __restrict__ agg,
                        const float* __restrict__ mass,
                        const float* __restrict__ centroids,
                        float* __restrict__ out,
                        float* __restrict__ bss)
{
  __shared__ float red[256];
  const int bk = blockIdx.x;           // b*K + k
  const int b  = bk >> 6;
  const int k  = bk & 63;
  const int tid = threadIdx.x;

  const float m = mass[bk];
  const size_t base = (size_t)bk * D_;
  const float v0 = agg[base + tid]       - m * centroids[(size_t)k * D_ + tid];
  const float v1 = agg[base + tid + 256] - m * centroids[(size_t)k * D_ + tid + 256];

  red[tid] = v0 * v0 + v1 * v1;
  __syncthreads();
  for (int s = 128; s > 0; s >>= 1) {
    if (tid < s) red[tid] += red[tid + s];
    __syncthreads();
  }
  const float ss  = red[0];
  const float inv = 1.0f / fmaxf(sqrtf(ss), EPSV);
  out[base + tid]       = v0 * inv;
  out[base + tid + 256] = v1 * inv;
  if (tid == 0) atomicAdd(&bss[b], ss * inv * inv);
}

// Final whole-descriptor L2 normalization per batch
__global__ void netvlad_scale_kernel(float* __restrict__ out,
                                     const float* __restrict__ bss)
{
  const int b = blockIdx.x;
  const float inv = 1.0f / fmaxf(sqrtf(bss[b]), EPSV);
  float* p = out + (size_t)b * K_ * D_;
  for (int i = threadIdx.x; i < K_ * D_; i += blockDim.x) p[i] *= inv;
}

extern "C" void kernel_launch(void* const* d_in, const int* in_sizes, int n_in,
                              void* d_out, int out_size, void* d_ws, size_t ws_size,
                              hipStream_t stream) {
  const float* x         = (const float*)d_in[0];   // [B, D, H, W]
  const float* centroids = (const float*)d_in[1];   // [K, D]
  const float* conv_w    = (const float*)d_in[2];   // [K, D]
  const float* conv_b    = (const float*)d_in[3];   // [K]
  float* out = (float*)d_out;                       // [B, K*D]

  float* agg  = (float*)d_ws;                       // [B,K,D]
  float* mass = agg + (size_t)B_ * K_ * D_;         // [B,K]
  float* bss  = mass + (size_t)B_ * K_;             // [B]

  const int zn = B_ * K_ * D_ + B_ * K_ + B_;
  zero_ws_kernel<<<(zn + 255) / 256, 256, 0, stream>>>(agg, zn);

  // Dynamic LDS: 2 x (512*68) + 64*66 + 64 floats  ~= 288.8 KB (<= 320 KB/WGP)
  const size_t shmem = (size_t)(2 * XBUF + K_ * AS + TN) * sizeof(float);
  netvlad_main_kernel<<<B_ * BLKS_PER_BATCH, 256, shmem, stream>>>(
      x, conv_w, conv_b, agg, mass);

  netvlad_finalize_kernel<<<B_ * K_, 256, 0, stream>>>(agg, mass, centroids, out, bss);
  netvlad_scale_kernel<<<B_, 256, 0, stream>>>(out, bss);
}